// JPEG_36696200577069
// MI455X (gfx1250) — compile-verified
//
#include <hip/hip_runtime.h>

// ---------------------------------------------------------------------------
// Fused differentiable-JPEG (quality=75) for MI455X / gfx1250, wave32.
// One pass: load RGB once -> YCbCr + 2x2 chroma pool -> per-8x8 DCT/quant/
// dequant/IDCT via V_WMMA_F32_16X16X4_F32 (4 blocks per 16x16 tile using a
// block-diagonal DCT operator) -> upsample + YCbCr->RGB -> store once.
// Memory-bound: ~201MB traffic => ~8.6us floor @ 23.3 TB/s.
// ---------------------------------------------------------------------------

typedef __attribute__((ext_vector_type(2))) float v2f;   // f32 A/B operand (16x4 / 4x16)
typedef __attribute__((ext_vector_type(8))) float v8f;   // f32 C/D operand (16x16)

__constant__ float c_qt_y[64] = {
  16,11,10,16,24,40,51,61,
  12,12,14,19,26,58,60,55,
  14,13,16,24,40,57,69,56,
  14,17,22,29,51,87,80,62,
  18,22,37,56,68,109,103,77,
  24,35,55,64,81,104,113,92,
  49,64,78,87,103,121,120,101,
  72,92,95,98,112,100,103,99};
__constant__ float c_qt_c[64] = {
  17,18,24,47,99,99,99,99,
  18,21,26,66,99,99,99,99,
  24,26,56,99,99,99,99,99,
  47,66,99,99,99,99,99,99,
  99,99,99,99,99,99,99,99,
  99,99,99,99,99,99,99,99,
  99,99,99,99,99,99,99,99,
  99,99,99,99,99,99,99,99};

// scaled_qt with QUALITY=75 -> s = 200 - 150 = 50, matching jnp.round (RNE).
__device__ __forceinline__ float scaled_qt(float q) {
  float v = q * 0.5f + 0.5f;            // (q*50 + 50) / 100
  float r = rintf(v);                   // round-half-even == jnp.round
  float t = v - r;
  float d = r + t * t * t;              // diff_round
  return fminf(fmaxf(d, 1.f), 255.f);
}

// 8x8 DCT matrix element C[u][x]
__device__ __forceinline__ float dct_c(int u, int x) {
  if (u == 0) return 0.35355339059327373f;                 // sqrt(1/8)
  return 0.5f * cosf((2.f * (float)x + 1.f) * (float)u * 0.19634954084936207f); // pi/16
}
// Block-diagonal diag(C, C): one 16x16 WMMA == 4 independent 8x8 transforms.
__device__ __forceinline__ float dm_elem(int r, int c) {
  if ((r < 8) != (c < 8)) return 0.f;
  return dct_c(r & 7, c & 7);
}

// acc(16x16) = A(16x16, 4 precomputed 16x4 chunks) * X^T, X row-major in LDS.
// f32 layouts (CDNA5 ISA 7.12.2): A 16x4 -> lane%16 = M, K = vgpr + 2*(lane>=16).
// B 4x16 symmetric: K = vgpr + 2*(lane>=16), N = lane%16.
// So B chunk k, vgpr v holds X^T[4k+v+2hi][n] = X[n][4k+v+2hi].
__device__ __forceinline__ v8f mm_pass(const v2f* a, const float* x,
                                       int stride, int n, int hi) {
  v8f acc = {};
#pragma unroll
  for (int k = 0; k < 4; ++k) {
    v2f bop;
    bop[0] = x[n * stride + 4 * k + 2 * hi + 0];
    bop[1] = x[n * stride + 4 * k + 2 * hi + 1];
    acc = __builtin_amdgcn_wmma_f32_16x16x4_f32(
        false, a[k], false, bop, (short)0, acc, false, false);
  }
  return acc;
}

// Scatter D-layout result (VGPR j -> row j + 8*hi, col n) row-major to LDS.
__device__ __forceinline__ void store_d(float* dst, int stride, v8f acc, int n, int hi) {
#pragma unroll
  for (int j = 0; j < 8; ++j) dst[(j + 8 * hi) * stride + n] = acc[j];
}

__device__ __forceinline__ void ycc_to_rgb(float y, float cb, float cr,
                                           float& r, float& g, float& b) {
  r = fminf(fmaxf(y + 1.402f * cr, 0.f), 255.f) * (1.f / 255.f);
  g = fminf(fmaxf(y - 0.344136f * cb - 0.714136f * cr, 0.f), 255.f) * (1.f / 255.f);
  b = fminf(fmaxf(y + 1.772f * cb, 0.f), 255.f) * (1.f / 255.f);
}

#define YSTR 65   // 64x64 Y tile, padded stride (bank-conflict friendly)
#define CSTR 33   // 32x32 chroma tiles, padded stride
#define SSTR 17   // per-wave 16x16 transpose scratch, padded stride

__global__ __launch_bounds__(256) void jpeg_fused_kernel(const float* __restrict__ img,
                                                         float* __restrict__ out) {
  __shared__ float sY [64 * YSTR];
  __shared__ float sCb[32 * CSTR];
  __shared__ float sCr[32 * CSTR];
  __shared__ float sScr[8 * 16 * SSTR];   // one 16x17 scratch per wave

  const int tid  = threadIdx.x;
  const int lane = tid & 31;
  const int wid  = tid >> 5;
  const size_t plane = 1024ull * 1024ull;
  const size_t base  = (size_t)blockIdx.z * 3 * plane;
  const int row0 = blockIdx.y * 64;
  const int col0 = blockIdx.x * 64;

  const float* pR = img + base;
  const float* pG = img + base + plane;
  const float* pB = img + base + 2 * plane;

  // ---------------- Phase A: load + RGB->YCbCr + chroma 2x2 avgpool --------
#pragma unroll
  for (int i = 0; i < 4; ++i) {
    int q  = tid + 256 * i;          // 2x2 quad index within 64x64 region
    int qr = q >> 5, qc = q & 31;
    size_t off = (size_t)(row0 + 2 * qr) * 1024 + (col0 + 2 * qc);
    float2 r0 = *(const float2*)(pR + off);
    float2 r1 = *(const float2*)(pR + off + 1024);
    float2 g0 = *(const float2*)(pG + off);
    float2 g1 = *(const float2*)(pG + off + 1024);
    float2 b0 = *(const float2*)(pB + off);
    float2 b1 = *(const float2*)(pB + off + 1024);
    r0.x *= 255.f; r0.y *= 255.f; r1.x *= 255.f; r1.y *= 255.f;
    g0.x *= 255.f; g0.y *= 255.f; g1.x *= 255.f; g1.y *= 255.f;
    b0.x *= 255.f; b0.y *= 255.f; b1.x *= 255.f; b1.y *= 255.f;

    float y00 = 0.299f*r0.x + 0.587f*g0.x + 0.114f*b0.x;
    float y01 = 0.299f*r0.y + 0.587f*g0.y + 0.114f*b0.y;
    float y10 = 0.299f*r1.x + 0.587f*g1.x + 0.114f*b1.x;
    float y11 = 0.299f*r1.y + 0.587f*g1.y + 0.114f*b1.y;
    // cb/cr computed without +128 (the +128 and the later -128 cancel)
    float cb = -0.168736f*(r0.x+r0.y+r1.x+r1.y)
               -0.331264f*(g0.x+g0.y+g1.x+g1.y)
               +0.5f     *(b0.x+b0.y+b1.x+b1.y);
    float cr =  0.5f     *(r0.x+r0.y+r1.x+r1.y)
               -0.418688f*(g0.x+g0.y+g1.x+g1.y)
               -0.081312f*(b0.x+b0.y+b1.x+b1.y);

    int yr = 2*qr, yc = 2*qc;
    sY[ yr     *YSTR + yc    ] = y00 - 128.f;
    sY[ yr     *YSTR + yc + 1] = y01 - 128.f;
    sY[(yr + 1)*YSTR + yc    ] = y10 - 128.f;
    sY[(yr + 1)*YSTR + yc + 1] = y11 - 128.f;
    sCb[qr*CSTR + qc] = 0.25f * cb;
    sCr[qr*CSTR + qc] = 0.25f * cr;
  }
  __syncthreads();

  // ---------------- Phase B: WMMA DCT -> quant/dequant -> IDCT -------------
  const int n  = lane & 15;
  const int hi = lane >> 4;
  v2f aF[4], aT[4];                      // Dm and Dm^T in A-operand layout
#pragma unroll
  for (int k = 0; k < 4; ++k) {
    aF[k][0] = dm_elem(n, 4*k + 2*hi);
    aF[k][1] = dm_elem(n, 4*k + 2*hi + 1);
    aT[k][0] = dm_elem(4*k + 2*hi,     n);
    aT[k][1] = dm_elem(4*k + 2*hi + 1, n);
  }
  // Quant value + reciprocal for D-layout position (u=j, v=lane&7).
  // Reciprocal hoists the expensive IEEE f32 divide out of the WMMA chain
  // (one real divide per value here, per workgroup, instead of per tile).
  float qYv[8], qCv[8], qYr[8], qCr[8];
#pragma unroll
  for (int j = 0; j < 8; ++j) {
    qYv[j] = scaled_qt(c_qt_y[j*8 + (lane & 7)]);
    qCv[j] = scaled_qt(c_qt_c[j*8 + (lane & 7)]);
    qYr[j] = 1.0f / qYv[j];
    qCr[j] = 1.0f / qCv[j];
  }
  float* scr = &sScr[wid * 16 * SSTR];

#pragma unroll
  for (int i = 0; i < 3; ++i) {
    int t = wid * 3 + i;                 // 24 tile-tasks over 8 waves
    bool isY = t < 16;
    float* tile; int stride;
    if (isY) {
      tile = &sY[((t >> 2) * 16) * YSTR + (t & 3) * 16]; stride = YSTR;
    } else {
      int u = t - 16;
      float* cbase = (u < 4) ? sCb : sCr;
      int ci = u & 3;
      tile = &cbase[((ci >> 1) * 16) * CSTR + (ci & 1) * 16]; stride = CSTR;
    }
    // d = Dm * X * Dm^T  via  U = Dm*X^T ; d = Dm*U^T (transpose thru scratch;
    // DS ops are in-order within a wave -> no barrier needed, scratch is wave-private)
    v8f acc = mm_pass(aF, tile, stride, n, hi);
    store_d(scr, SSTR, acc, n, hi);
    acc = mm_pass(aF, scr, SSTR, n, hi);
    // quant/dequant: diff_round(d * (1/q)) * q, index (u=j, v=lane&7)
#pragma unroll
    for (int j = 0; j < 8; ++j) {
      float qq = isY ? qYv[j] : qCv[j];
      float qi = isY ? qYr[j] : qCr[j];
      float x = acc[j] * qi;
      float r = rintf(x);
      float tt = x - r;
      acc[j] = (r + tt * tt * tt) * qq;
    }
    // rec = Dm^T * e * Dm  via  W = Dm^T*e^T ; rec = Dm^T*W^T
    store_d(scr, SSTR, acc, n, hi);
    acc = mm_pass(aT, scr, SSTR, n, hi);
    store_d(scr, SSTR, acc, n, hi);
    acc = mm_pass(aT, scr, SSTR, n, hi);
    store_d(tile, stride, acc, n, hi);   // write reconstruction in place
  }
  __syncthreads();

  // ---------------- Phase C: upsample + YCbCr->RGB + store -----------------
  float* oR = out + base;
  float* oG = out + base + plane;
  float* oB = out + base + 2 * plane;
#pragma unroll
  for (int i = 0; i < 4; ++i) {
    int q  = tid + 256 * i;
    int qr = q >> 5, qc = q & 31;
    int yr = 2*qr, yc = 2*qc;
    float cbv = sCb[qr*CSTR + qc];       // rec chroma; (+128 - 128) cancels
    float crv = sCr[qr*CSTR + qc];
    float y00 = sY[ yr     *YSTR + yc    ] + 128.f;
    float y01 = sY[ yr     *YSTR + yc + 1] + 128.f;
    float y10 = sY[(yr + 1)*YSTR + yc    ] + 128.f;
    float y11 = sY[(yr + 1)*YSTR + yc + 1] + 128.f;
    float r00,g00,b00, r01,g01,b01, r10,g10,b10, r11,g11,b11;
    ycc_to_rgb(y00, cbv, crv, r00, g00, b00);
    ycc_to_rgb(y01, cbv, crv, r01, g01, b01);
    ycc_to_rgb(y10, cbv, crv, r10, g10, b10);
    ycc_to_rgb(y11, cbv, crv, r11, g11, b11);
    size_t off = (size_t)(row0 + yr) * 1024 + (col0 + yc);
    *(float2*)(oR + off)        = make_float2(r00, r01);
    *(float2*)(oR + off + 1024) = make_float2(r10, r11);
    *(float2*)(oG + off)        = make_float2(g00, g01);
    *(float2*)(oG + off + 1024) = make_float2(g10, g11);
    *(float2*)(oB + off)        = make_float2(b00, b01);
    *(float2*)(oB + off + 1024) = make_float2(b10, b11);
  }
}

extern "C" void kernel_launch(void* const* d_in, const int* in_sizes, int n_in,
                              void* d_out, int out_size, void* d_ws, size_t ws_size,
                              hipStream_t stream) {
  (void)n_in; (void)out_size; (void)d_ws; (void)ws_size;
  const float* img = (const float*)d_in[0];
  float* out = (float*)d_out;
  int B = in_sizes[0] / (3 * 1024 * 1024);   // 8 for the reference setup
  dim3 grid(1024 / 64, 1024 / 64, B);
  jpeg_fused_kernel<<<grid, dim3(256), 0, stream>>>(img, out);
}